// FractalAnisotropicDiffusion_63814624084055
// MI455X (gfx1250) — compile-verified
//
#include <hip/hip_runtime.h>
#include <hip/hip_bf16.h>

#define B_    16
#define H_    512
#define W_    512
#define HW_   (H_ * W_)
#define NTOT_ (B_ * HW_)
#define RAD   10

typedef __attribute__((ext_vector_type(2))) float v2f;
typedef __attribute__((ext_vector_type(8))) float v8f;

__device__ __forceinline__ int refl(int i, int n) {
    i = (i < 0) ? -i : i;
    return (i >= n) ? (2 * n - 2 - i) : i;
}

__device__ __forceinline__ float sigm(float x) { return 1.0f / (1.0f + expf(-x)); }

// ---------------------------------------------------------------------------
// Prep: derived scalar params + normalized 21-tap Gaussian, zero-padded into a
// 61-entry table wext so wext[20 + t] = w[t] for t in [0,20], else 0.
// P[0..7]=alpha,lam,beta,xi,eta,nu,gamma,omega ; P[16..76]=wext ; P[96..111]=er_max bits
// ---------------------------------------------------------------------------
__global__ void prep_kernel(const float* a_raw, const float* l_raw, const float* ls,
                            const float* lb, const float* lx, const float* e_raw,
                            const float* n_raw, const float* lg, const float* o_raw,
                            float* P) {
    if (threadIdx.x != 0) return;
    float alpha = 0.6f  + 1.4f  * sigm(*a_raw);
    float lam   = 0.01f + 0.19f * sigm(*l_raw);
    float sigma = 0.5f  + 4.5f  * sigm(*ls);
    float beta  = 1.0f  + 4.0f  * sigm(*lb);
    float xi    = 1.0f  + 4.0f  * sigm(*lx);
    float eta   = sigm(*e_raw);
    float nu    = sigm(*n_raw);
    float gamma = 1.0f  + 3.0f  * sigm(*lg);
    float omega = sigm(*o_raw);
    P[0] = alpha; P[1] = lam; P[2] = beta; P[3] = xi;
    P[4] = eta;   P[5] = nu;  P[6] = gamma; P[7] = omega;

    float wv[2 * RAD + 1];
    float s = 0.f;
    float inv2s2 = 1.0f / (2.0f * sigma * sigma + 1e-8f);
    for (int i = 0; i < 2 * RAD + 1; ++i) {
        float d = (float)(i - RAD);
        wv[i] = expf(-d * d * inv2s2);
        s += wv[i];
    }
    float inv = 1.0f / (s + 1e-8f);
    for (int j = 0; j < 61; ++j) P[16 + j] = 0.f;
    for (int i = 0; i < 2 * RAD + 1; ++i) P[16 + 20 + i] = wv[i] * inv;
}

// ---------------------------------------------------------------------------
// Horizontal blur via WMMA: per wave, one 16x16 tile.
//   D[r][c] = sum_{j=0..35} In[y0+r][x0-10+j] * M[j][c],  M[j][c] = w[j-c]
// 9 chunks of K=4 -> 9x V_WMMA_F32_16X16X4_F32.
// A layout: m = lane&15 ; VGPR p holds K = (lane>>4)*2 + p   (ISA 7.12.2)
// B layout: n = lane&15 ; VGPR p holds K = (lane>>4)*2 + p
// D layout: n = lane&15 ; VGPR v holds M = v + 8*(lane>>4)
// ---------------------------------------------------------------------------
__global__ __launch_bounds__(256) void blur_h_kernel(const float* __restrict__ src,
                                                     float* __restrict__ dst,
                                                     const float* __restrict__ P) {
    const float* wext = P + 16;
    int wave = blockIdx.x * (blockDim.x >> 5) + (threadIdx.x >> 5);
    int lane = threadIdx.x & 31;
    const int tilesX = W_ / 16, tilesY = H_ / 16;
    int b  = wave / (tilesX * tilesY);
    int t  = wave % (tilesX * tilesY);
    int y0 = (t / tilesX) * 16;
    int x0 = (t % tilesX) * 16;
    int r  = lane & 15;        // matrix row (A) / column (B,D)
    int kh = lane >> 4;        // lane-half selects K pairs

    const float* s = src + (size_t)b * HW_;
    const float* srow = s + (size_t)(y0 + r) * W_;
    v8f acc = {};
#pragma unroll
    for (int ch = 0; ch < 9; ++ch) {
        int k = 4 * ch + 2 * kh;
        v2f a, bv;
        a.x  = srow[refl(x0 - RAD + k,     W_)];
        a.y  = srow[refl(x0 - RAD + k + 1, W_)];
        bv.x = wext[20 + k     - r];
        bv.y = wext[20 + k + 1 - r];
        acc = __builtin_amdgcn_wmma_f32_16x16x4_f32(false, a, false, bv,
                                                    (short)0, acc, false, false);
    }
    float* d = dst + (size_t)b * HW_;
    union { v8f v; float f[8]; } o; o.v = acc;
#pragma unroll
    for (int v = 0; v < 8; ++v)
        d[(size_t)(y0 + v + 8 * kh) * W_ + x0 + r] = o.f[v];
}

// ---------------------------------------------------------------------------
// Vertical blur via WMMA: D[r][c] = sum_j w[j-r] * In[y0-10+j][x0+c]
// A = banded weight matrix (16x36), B = input columns (36x16). Coalesced B loads.
// ---------------------------------------------------------------------------
__global__ __launch_bounds__(256) void blur_v_kernel(const float* __restrict__ src,
                                                     float* __restrict__ dst,
                                                     const float* __restrict__ P) {
    const float* wext = P + 16;
    int wave = blockIdx.x * (blockDim.x >> 5) + (threadIdx.x >> 5);
    int lane = threadIdx.x & 31;
    const int tilesX = W_ / 16, tilesY = H_ / 16;
    int b  = wave / (tilesX * tilesY);
    int t  = wave % (tilesX * tilesY);
    int y0 = (t / tilesX) * 16;
    int x0 = (t % tilesX) * 16;
    int r  = lane & 15;
    int kh = lane >> 4;

    const float* s = src + (size_t)b * HW_;
    v8f acc = {};
#pragma unroll
    for (int ch = 0; ch < 9; ++ch) {
        int k = 4 * ch + 2 * kh;
        v2f a, bv;
        a.x  = wext[20 + k     - r];
        a.y  = wext[20 + k + 1 - r];
        bv.x = s[(size_t)refl(y0 - RAD + k,     H_) * W_ + x0 + r];
        bv.y = s[(size_t)refl(y0 - RAD + k + 1, H_) * W_ + x0 + r];
        acc = __builtin_amdgcn_wmma_f32_16x16x4_f32(false, a, false, bv,
                                                    (short)0, acc, false, false);
    }
    float* d = dst + (size_t)b * HW_;
    union { v8f v; float f[8]; } o; o.v = acc;
#pragma unroll
    for (int v = 0; v < 8; ++v)
        d[(size_t)(y0 + v + 8 * kh) * W_ + x0 + r] = o.f[v];
}

// ---------------------------------------------------------------------------
// phi_f = min(beta*sqrt(xi/(eta*|grad u_sigma|^2 + 1e-6)), 10) * clamp(1-omega*lfd,0,1)
// ---------------------------------------------------------------------------
__global__ __launch_bounds__(256) void phi_kernel(const float* __restrict__ usig,
                                                  const float* __restrict__ lfd,
                                                  float* __restrict__ out,
                                                  const float* __restrict__ P) {
    size_t i = (size_t)blockIdx.x * blockDim.x + threadIdx.x;
    if (i >= NTOT_) return;
    int x = (int)(i % W_);
    int y = (int)((i / W_) % H_);
    size_t base = i - (size_t)(y * W_ + x);
    float gxs = 0.5f * (usig[base + (size_t)y * W_ + refl(x + 1, W_)]
                      - usig[base + (size_t)y * W_ + refl(x - 1, W_)]);
    float gys = 0.5f * (usig[base + (size_t)refl(y + 1, H_) * W_ + x]
                      - usig[base + (size_t)refl(y - 1, H_) * W_ + x]);
    float gm2 = gxs * gxs + gys * gys + 1e-8f;
    float beta = P[2], xi = P[3], eta = P[4], omega = P[7];
    float phi = fminf(beta * sqrtf(xi / (eta * gm2 + 1e-6f)), 10.0f);
    float fw  = fminf(fmaxf(1.0f - omega * lfd[i], 0.0f), 1.0f);
    out[i] = phi * fw;
}

// ---------------------------------------------------------------------------
// Diffusion update (radius-1 stencil on u and phi_f)
// ---------------------------------------------------------------------------
__global__ __launch_bounds__(256) void update_kernel(const float* __restrict__ u_in,
                                                     const float* __restrict__ phi,
                                                     const float* __restrict__ u0,
                                                     float* __restrict__ u_out,
                                                     const float* __restrict__ P) {
    size_t i = (size_t)blockIdx.x * blockDim.x + threadIdx.x;
    if (i >= NTOT_) return;
    int x = (int)(i % W_);
    int y = (int)((i / W_) % H_);
    size_t base = i - (size_t)(y * W_ + x);
    int ym = refl(y - 1, H_), yp = refl(y + 1, H_);
    int xm = refl(x - 1, W_), xp = refl(x + 1, W_);

    float u  = u_in[i];
    float un = u_in[base + (size_t)ym * W_ + x];
    float us = u_in[base + (size_t)yp * W_ + x];
    float uw = u_in[base + (size_t)y  * W_ + xm];
    float ue = u_in[base + (size_t)y  * W_ + xp];

    float gy = 0.5f * (us - un);
    float gx = 0.5f * (ue - uw);
    float gm  = sqrtf(gx * gx + gy * gy + 1e-8f);
    float lap = un + us + ue + uw - 4.0f * u;
    float s   = gm * fabsf(lap);

    float alpha = P[0], lam = P[1], nu = P[5], gamma = P[6];
    float psi = 1e-4f * sqrtf(nu * s * s * s + gamma);

    float pn = phi[base + (size_t)ym * W_ + x];
    float ps = phi[base + (size_t)yp * W_ + x];
    float pw = phi[base + (size_t)y  * W_ + xm];
    float pe = phi[base + (size_t)y  * W_ + xp];
    float div = pn * (un - u) + ps * (us - u) + pe * (ue - u) + pw * (uw - u);

    float du = alpha * psi * div - lam * (u - u0[i]);
    float r  = u + 0.1f * du;
    u_out[i] = fminf(fmaxf(r, 0.0f), 1.0f);
}

// ---------------------------------------------------------------------------
// Per-image max of |image - u| (uint-bit atomicMax; valid for non-negative f32)
// ---------------------------------------------------------------------------
__global__ void initmax_kernel(unsigned int* mx) {
    if (threadIdx.x < B_) mx[threadIdx.x] = 0u;
}

__global__ __launch_bounds__(256) void ermax_kernel(const float* __restrict__ img,
                                                    const float* u,
                                                    unsigned int* __restrict__ mx) {
    __shared__ float red[256];
    int b     = blockIdx.x >> 5;   // 32 chunks per image
    int chunk = blockIdx.x & 31;
    size_t base = (size_t)b * HW_;
    const int CH = HW_ / 32;
    float m = 0.f;
    for (int j = chunk * CH + threadIdx.x; j < (chunk + 1) * CH; j += 256)
        m = fmaxf(m, fabsf(img[base + j] - u[base + j]));
    red[threadIdx.x] = m;
    __syncthreads();
    for (int s = 128; s > 0; s >>= 1) {
        if (threadIdx.x < s) red[threadIdx.x] = fmaxf(red[threadIdx.x], red[threadIdx.x + s]);
        __syncthreads();
    }
    if (threadIdx.x == 0) atomicMax(&mx[b], __float_as_uint(red[0]));
}

// out[0..N) = u ; out[N..2N) = |img-u|/(max+1e-8).  u aliases out+N: each thread
// reads its own u[i] before writing out[N+i], so in-place is safe.
__global__ __launch_bounds__(256) void finalize_kernel(const float* __restrict__ img,
                                                       const float* u,
                                                       const unsigned int* __restrict__ mx,
                                                       float* out) {
    size_t i = (size_t)blockIdx.x * blockDim.x + threadIdx.x;
    if (i >= NTOT_) return;
    int b = (int)(i / HW_);
    float uv = u[i];
    float m  = __uint_as_float(mx[b]);
    float er = fabsf(img[i] - uv) / (m + 1e-8f);
    out[i]         = uv;
    out[NTOT_ + i] = er;
}

// ---------------------------------------------------------------------------
extern "C" void kernel_launch(void* const* d_in, const int* in_sizes, int n_in,
                              void* d_out, int out_size, void* d_ws, size_t ws_size,
                              hipStream_t stream) {
    const float* image   = (const float*)d_in[0];
    const float* lfd     = (const float*)d_in[1];
    const float* p_alpha = (const float*)d_in[2];
    const float* p_lam   = (const float*)d_in[3];
    const float* p_lsig  = (const float*)d_in[4];
    const float* p_lbeta = (const float*)d_in[5];
    const float* p_lxi   = (const float*)d_in[6];
    const float* p_eta   = (const float*)d_in[7];
    const float* p_nu    = (const float*)d_in[8];
    const float* p_lgam  = (const float*)d_in[9];
    const float* p_omega = (const float*)d_in[10];

    float* P   = (float*)d_ws;          // 128-float param block
    float* F0  = P + 128;               // one field in scratch
    float* out = (float*)d_out;
    float* F1  = out;                   // d_out first half as scratch field
    float* F2  = out + NTOT_;           // d_out second half as scratch field

    hipMemcpyAsync(F0, image, (size_t)NTOT_ * sizeof(float),
                   hipMemcpyDeviceToDevice, stream);
    prep_kernel<<<1, 32, 0, stream>>>(p_alpha, p_lam, p_lsig, p_lbeta, p_lxi,
                                      p_eta, p_nu, p_lgam, p_omega, P);

    const int tiles      = B_ * (H_ / 16) * (W_ / 16);  // 16384
    const int blurBlocks = tiles / 8;                   // 8 waves/block
    const int ptBlocks   = NTOT_ / 256;

    float* u     = F0;
    float* other = F2;
    for (int step = 0; step < 5; ++step) {
        blur_h_kernel<<<blurBlocks, 256, 0, stream>>>(u, F1, P);       // u -> htmp
        blur_v_kernel<<<blurBlocks, 256, 0, stream>>>(F1, other, P);   // htmp -> u_sigma
        phi_kernel<<<ptBlocks, 256, 0, stream>>>(other, lfd, F1, P);   // u_sigma -> phi_f
        update_kernel<<<ptBlocks, 256, 0, stream>>>(u, F1, image, other, P);
        float* t = u; u = other; other = t;
    }
    // After 5 steps u == F2 (== out + NTOT_)

    unsigned int* mx = (unsigned int*)(P + 96);
    initmax_kernel<<<1, 32, 0, stream>>>(mx);
    ermax_kernel<<<B_ * 32, 256, 0, stream>>>(image, u, mx);
    finalize_kernel<<<ptBlocks, 256, 0, stream>>>(image, u, mx, out);
}